// GATConv_60842506715227
// MI455X (gfx1250) — compile-verified
//
#include <hip/hip_runtime.h>
#include <hip/hip_bf16.h>

#define N_NODES 4096
#define H_HEADS 8
#define FIN     256
#define FOUT    64
#define DEG     32
#define E_EDGES (N_NODES * DEG)
#define LRELU_ALPHA 0.2f

typedef __attribute__((ext_vector_type(2))) float v2f;
typedef __attribute__((ext_vector_type(8))) float v8f;
typedef __attribute__((ext_vector_type(4))) int   v4i;

#if defined(__gfx1250__)
typedef __attribute__((address_space(1))) v4i* as1_v4i_ptr;   // global
typedef __attribute__((address_space(3))) v4i* as3_v4i_ptr;   // LDS
#endif

// ---------------------------------------------------------------------------
// Kernel 1: per-head GEMM  h[head,n,o] = sum_i x[n,i] * W[head,i,o]
//
// Block = 256 threads (8 waves), one head per blockIdx.y.
// W[head] (256x64 f32 = 64 KB) is staged in LDS once per block, via
// GLOBAL_LOAD_ASYNC_TO_LDS_B128 (ASYNCcnt-tracked) when available.
// Each wave then computes a 16x64 output strip: for each K step one
// contiguous b64 A load feeds FOUR v_wmma_f32_16x16x4_f32 ops whose B
// operands come from LDS. 256 WMMAs / wave, A traffic amortized 4x.
// ---------------------------------------------------------------------------
__global__ __launch_bounds__(256) void gat_gemm_wmma(
    const float* __restrict__ x,      // (N, FIN)
    const float* __restrict__ w,      // (H, FIN, FOUT)
    float* __restrict__ hbuf)         // (H, N, FOUT)
{
  __shared__ __align__(16) float wlds[FIN * FOUT];   // 64 KB

  const int head = blockIdx.y;
  const int tid  = threadIdx.x;
  const int lane = tid & 31;
  const int wave = tid >> 5;
  const int mt   = blockIdx.x * 8 + wave;            // 0..255 (M tile)
  const int half = lane >> 4;                        // 0: lanes 0-15, 1: 16-31
  const int l16  = lane & 15;

  // ---- stage W[head] into LDS -------------------------------------------
  const float* wsrc = w + (size_t)head * FIN * FOUT;
#if defined(__gfx1250__) && __has_builtin(__builtin_amdgcn_global_load_async_to_lds_b128)
  #pragma unroll
  for (int i = 0; i < (FIN * FOUT) / (256 * 4); ++i) {       // 16 x b128 / thread
    const int gidx = (i * 256 + tid) * 4;
    __builtin_amdgcn_global_load_async_to_lds_b128(
        (as1_v4i_ptr)(wsrc + gidx),
        (as3_v4i_ptr)&wlds[gidx],
        /*offset=*/0, /*cpol=*/0);
  }
  asm volatile("s_wait_asynccnt 0x0" ::: "memory");          // drain ASYNCcnt
#else
  #pragma unroll
  for (int i = 0; i < (FIN * FOUT) / (256 * 4); ++i) {
    const int gidx = (i * 256 + tid) * 4;
    *(float4*)&wlds[gidx] = *(const float4*)(wsrc + gidx);
  }
#endif
  __syncthreads();

  // ---- 16x64 strip per wave ---------------------------------------------
  // A-matrix layout (32-bit 16x4): lane row = M; lanes 0-15 K={k,k+1},
  // lanes 16-31 K={k+2,k+3}  -> one contiguous float2 per lane.
  const float* xrow = x + (size_t)(mt * 16 + l16) * FIN + half * 2;

  v8f c0 = {}, c1 = {}, c2 = {}, c3 = {};
  #pragma unroll 2
  for (int k = 0; k < FIN; k += 4) {
    const v2f a = *(const v2f*)(xrow + k);

    // B-matrix layout (4x16): lane = N column; VGPR0/1 = K rows kb, kb+1.
    const int kb = k + half * 2;
    const float* bp = &wlds[kb * FOUT + l16];
    v2f b0, b1, b2, b3;
    b0.x = bp[0];   b0.y = bp[FOUT + 0];
    b1.x = bp[16];  b1.y = bp[FOUT + 16];
    b2.x = bp[32];  b2.y = bp[FOUT + 32];
    b3.x = bp[48];  b3.y = bp[FOUT + 48];

    c0 = __builtin_amdgcn_wmma_f32_16x16x4_f32(false, a, false, b0, (short)0, c0, false, false);
    c1 = __builtin_amdgcn_wmma_f32_16x16x4_f32(false, a, false, b1, (short)0, c1, false, false);
    c2 = __builtin_amdgcn_wmma_f32_16x16x4_f32(false, a, false, b2, (short)0, c2, false, false);
    c3 = __builtin_amdgcn_wmma_f32_16x16x4_f32(false, a, false, b3, (short)0, c3, false, false);
  }

  // C/D layout: VGPR r -> row mt*16 + r + 8*half, col = ntile*16 + l16.
  float* crow = hbuf + (size_t)head * N_NODES * FOUT
                     + (size_t)(mt * 16 + half * 8) * FOUT + l16;
  #pragma unroll
  for (int r = 0; r < 8; ++r) {
    crow[(size_t)r * FOUT +  0] = c0[r];
    crow[(size_t)r * FOUT + 16] = c1[r];
    crow[(size_t)r * FOUT + 32] = c2[r];
    crow[(size_t)r * FOUT + 48] = c3[r];
  }
}

// ---------------------------------------------------------------------------
// Kernel 2: per-(head,node) attention score halves:
//   s_src[h,n] = h[h,n,:] . a[h, :64],   s_tgt[h,n] = h[h,n,:] . a[h, 64:]
// One wave per (h,n); 32 lanes x 2 elements, shfl_xor tree reduction.
// ---------------------------------------------------------------------------
__global__ __launch_bounds__(256) void gat_scores(
    const float* __restrict__ hbuf,   // (H, N, FOUT)
    const float* __restrict__ a,      // (H, 2*FOUT, 1)
    float* __restrict__ s_src,        // (H*N)
    float* __restrict__ s_tgt)        // (H*N)
{
  const int lane = threadIdx.x & 31;
  const int wid  = (int)((blockIdx.x * blockDim.x + threadIdx.x) >> 5); // 0..H*N-1
  const int head = wid >> 12;                   // / 4096
  const int n    = wid & (N_NODES - 1);

  const float* hrow = hbuf + ((size_t)head * N_NODES + n) * FOUT;
  const float* av   = a + head * 2 * FOUT;

  const float v0 = hrow[lane];
  const float v1 = hrow[lane + 32];
  float ss = v0 * av[lane]        + v1 * av[lane + 32];
  float st = v0 * av[FOUT + lane] + v1 * av[FOUT + lane + 32];
  #pragma unroll
  for (int m = 16; m >= 1; m >>= 1) {
    ss += __shfl_xor(ss, m, 32);
    st += __shfl_xor(st, m, 32);
  }
  if (lane == 0) {
    s_src[wid] = ss;
    s_tgt[wid] = st;
  }
}

// ---------------------------------------------------------------------------
// Kernel 3: per node n (block), per head (wave):
//   score_j = leaky_relu(s_src[h,n] + s_tgt[h,tgt_j]) * edge_attr[j]
//   alpha   = softmax_32(score)           (dense-NxN softmax collapses to this)
//   out[n, h*64+o] = sum_j alpha_j * h[h, tgt_j, o] + bias[h,o]
// Lane j owns edge j for the softmax; lane l owns channels {l, l+32} for the
// aggregation (coalesced loads of h rows).
// ---------------------------------------------------------------------------
__global__ __launch_bounds__(256) void gat_aggregate(
    const int*   __restrict__ tgt,        // (E)  targets, sorted by src
    const float* __restrict__ edge_attr,  // (E)
    const float* __restrict__ hbuf,       // (H, N, FOUT)
    const float* __restrict__ s_src,      // (H*N)
    const float* __restrict__ s_tgt,      // (H*N)
    const float* __restrict__ bias,       // (H, 1, FOUT)
    float* __restrict__ out)              // (N, H*FOUT)
{
  __shared__ int   sh_tgt[DEG];
  __shared__ float sh_ea[DEG];
  __shared__ float sh_w[H_HEADS][DEG];

  const int n    = blockIdx.x;
  const int lane = threadIdx.x & 31;
  const int head = threadIdx.x >> 5;

  if (threadIdx.x < DEG) {
    sh_tgt[threadIdx.x] = tgt[n * DEG + threadIdx.x];
    sh_ea[threadIdx.x]  = edge_attr[n * DEG + threadIdx.x];
  }
  __syncthreads();

  // lane j: score of edge j
  const int t = sh_tgt[lane];
  float sc = s_src[head * N_NODES + n] + s_tgt[head * N_NODES + t];
  sc = (sc > 0.0f) ? sc : LRELU_ALPHA * sc;
  sc *= sh_ea[lane];

  // wave32 softmax over the 32 edge scores
  float mx = sc;
  #pragma unroll
  for (int m = 16; m >= 1; m >>= 1) mx = fmaxf(mx, __shfl_xor(mx, m, 32));
  const float ex = __expf(sc - mx);
  float sum = ex;
  #pragma unroll
  for (int m = 16; m >= 1; m >>= 1) sum += __shfl_xor(sum, m, 32);
  sh_w[head][lane] = ex / sum;
  __syncthreads();

  // lane l accumulates output channels l and l+32
  float acc0 = 0.0f, acc1 = 0.0f;
  #pragma unroll 4
  for (int j = 0; j < DEG; ++j) {
    const float wj = sh_w[head][j];
    const float* hp = hbuf + ((size_t)head * N_NODES + sh_tgt[j]) * FOUT;
    acc0 += wj * hp[lane];
    acc1 += wj * hp[lane + 32];
  }

  float* orow = out + (size_t)n * (H_HEADS * FOUT) + head * FOUT;
  orow[lane]      = acc0 + bias[head * FOUT + lane];
  orow[lane + 32] = acc1 + bias[head * FOUT + lane + 32];
}

// ---------------------------------------------------------------------------
extern "C" void kernel_launch(void* const* d_in, const int* in_sizes, int n_in,
                              void* d_out, int out_size, void* d_ws, size_t ws_size,
                              hipStream_t stream) {
  const float* x         = (const float*)d_in[0];
  const int*   edge_list = (const int*)  d_in[1];
  const float* edge_attr = (const float*)d_in[2];
  const float* weight    = (const float*)d_in[3];
  const float* a         = (const float*)d_in[4];
  const float* bias      = (const float*)d_in[5];
  float*       out       = (float*)d_out;

  // workspace: h (H*N*FOUT floats) | s_src (H*N) | s_tgt (H*N)
  float* hbuf  = (float*)d_ws;
  float* s_src = hbuf + (size_t)H_HEADS * N_NODES * FOUT;
  float* s_tgt = s_src + (size_t)H_HEADS * N_NODES;

  const int* tgt = edge_list + E_EDGES;   // edge_list[1, :]

  // 1) h = x @ W per head (WMMA f32 16x16x4, W staged in LDS via async DMA)
  gat_gemm_wmma<<<dim3(32, H_HEADS), 256, 0, stream>>>(x, weight, hbuf);
  // 2) attention score halves
  gat_scores<<<(H_HEADS * N_NODES) / 8, 256, 0, stream>>>(hbuf, a, s_src, s_tgt);
  // 3) softmax over 32 neighbors + weighted aggregation + bias
  gat_aggregate<<<N_NODES, 256, 0, stream>>>(tgt, edge_attr, hbuf, s_src, s_tgt,
                                             bias, out);
}